// RRN_55439437856890
// MI455X (gfx1250) — compile-verified
//
#include <hip/hip_runtime.h>
#include <hip/hip_bf16.h>
#include <math.h>

// ---------------------------------------------------------------------------
// Problem constants (match reference)
// ---------------------------------------------------------------------------
#define NN   10000      // n_units
#define EE   320000     // n_edges
#define DD   128        // dim_input == dim_hidden
#define FF   256        // f hidden dim
#define MM   128        // message dim
#define OHID 128
#define OOUT 64

typedef __attribute__((ext_vector_type(2))) float v2f;
typedef __attribute__((ext_vector_type(8))) float v8f;

// Full-precision matrix op: D(16x16,f32) = A(16x4,f32) * B(4x16,f32) + C
// 8-arg form: (neg_a, A, neg_b, B, c_mod, C, reuse_a, reuse_b)
#define WMMA_F32X4(a, b, c) \
  __builtin_amdgcn_wmma_f32_16x16x4_f32(false, (a), false, (b), (short)0, (c), false, false)

__device__ __forceinline__ float sigmoid_(float v) {
  return 1.0f / (1.0f + __expf(-v));
}

__device__ __forceinline__ v8f splat8(float v) {
  v8f r;
#pragma unroll
  for (int i = 0; i < 8; ++i) r[i] = v;
  return r;
}

// ---------------------------------------------------------------------------
// Kernel 1: edge message MLP + scatter-add.  One wave = 32 edges (2 M-tiles
// sharing every B fragment -> halves the Wf1/Wf2 L2 stream per FLOP).
// Layer1 [32,256]@[256,256]+relu runs as two passes of 8 N-tiles (keeps the
// accumulator footprint at 128 VGPRs); layer2 [32,256]@[256,128]; messages
// scatter-added into summsg[dst] with hardware f32 L2 atomics.
// Fragment layouts (ISA 7.12.2):
//   A (16x4 f32): lane<16 -> A[lm][k..k+1], lane>=16 -> A[lm][k+2..k+3]
//   B (4x16 f32): vgpr0 = B[k+2*half][col], vgpr1 = B[k+2*half+1][col]
//   C tile:       vgpr i -> row (i + 8*half), col = n*16 + lm
// ---------------------------------------------------------------------------
__global__ __launch_bounds__(32) void rrn_edge_mlp(
    const float* __restrict__ hidden, const int* __restrict__ src,
    const int* __restrict__ dst, const float* __restrict__ Wf1,
    const float* __restrict__ bf1, const float* __restrict__ Wf2,
    const float* __restrict__ bf2, float* __restrict__ summsg) {
  constexpr int LDK = 260;                 // 256 + 4 pad -> conflict-free transpose
  __shared__ float h1[32 * LDK];           // layer-1 activations (33.3 KB)
  __shared__ int dstl[32];

  const int lane = threadIdx.x & 31;
  const int half = lane >> 4;
  const int lm = lane & 15;
  const int e0 = blockIdx.x * 32;
  if (e0 >= EE) return;

  // two edge rows per lane slot: M-tile 0 = edges e0..e0+15, tile 1 = +16
  const int ea = e0 + lm;
  const int eb = e0 + 16 + lm;
  const int sa = src[ea], da = dst[ea];
  const int sb = src[eb], db = dst[eb];
  dstl[lm] = da;
  dstl[16 + lm] = db;

  const float* As0 = hidden + (size_t)sa * DD;  // tile0, A cols   0..127
  const float* Ad0 = hidden + (size_t)da * DD;  // tile0, A cols 128..255
  const float* As1 = hidden + (size_t)sb * DD;  // tile1, A cols   0..127
  const float* Ad1 = hidden + (size_t)db * DD;  // tile1, A cols 128..255

  // ---- layer 1: two passes over the 16 N-tiles ----
#pragma unroll 1
  for (int p = 0; p < 2; ++p) {
    v8f acc[2][8];
#pragma unroll
    for (int n = 0; n < 8; ++n) {
      const float bv = bf1[(p * 8 + n) * 16 + lm];
      acc[0][n] = splat8(bv);
      acc[1][n] = splat8(bv);
    }
    // K = 0..127 : src endpoint
#pragma unroll 1
    for (int k = 0; k < 128; k += 4) {
      const int kk = k + 2 * half;
      v2f a0, a1;
      a0.x = As0[kk]; a0.y = As0[kk + 1];
      a1.x = As1[kk]; a1.y = As1[kk + 1];
      const float* wr = Wf1 + (size_t)kk * FF + p * 128;
#pragma unroll
      for (int n = 0; n < 8; ++n) {
        v2f b;
        b.x = wr[n * 16 + lm];
        b.y = wr[FF + n * 16 + lm];
        acc[0][n] = WMMA_F32X4(a0, b, acc[0][n]);
        acc[1][n] = WMMA_F32X4(a1, b, acc[1][n]);
      }
    }
    // K = 128..255 : dst endpoint
#pragma unroll 1
    for (int k = 0; k < 128; k += 4) {
      const int kk = k + 2 * half;
      v2f a0, a1;
      a0.x = Ad0[kk]; a0.y = Ad0[kk + 1];
      a1.x = Ad1[kk]; a1.y = Ad1[kk + 1];
      const float* wr = Wf1 + (size_t)(128 + kk) * FF + p * 128;
#pragma unroll
      for (int n = 0; n < 8; ++n) {
        v2f b;
        b.x = wr[n * 16 + lm];
        b.y = wr[FF + n * 16 + lm];
        acc[0][n] = WMMA_F32X4(a0, b, acc[0][n]);
        acc[1][n] = WMMA_F32X4(a1, b, acc[1][n]);
      }
    }
    // relu + C-layout -> A-layout transpose through LDS (same-wave DS in order)
#pragma unroll
    for (int m = 0; m < 2; ++m) {
#pragma unroll
      for (int n = 0; n < 8; ++n) {
#pragma unroll
        for (int i = 0; i < 8; ++i) {
          h1[(m * 16 + i + 8 * half) * LDK + (p * 8 + n) * 16 + lm] =
              fmaxf(acc[m][n][i], 0.0f);
        }
      }
    }
  }

  // ---- layer 2: [32,256] @ Wf2[256,128] ----
  v8f acc2[2][8];
#pragma unroll
  for (int n = 0; n < 8; ++n) {
    const float bv = bf2[n * 16 + lm];
    acc2[0][n] = splat8(bv);
    acc2[1][n] = splat8(bv);
  }
#pragma unroll 1
  for (int k = 0; k < 256; k += 4) {
    const int kk = k + 2 * half;
    v2f a0, a1;
    a0.x = h1[lm * LDK + kk];
    a0.y = h1[lm * LDK + kk + 1];
    a1.x = h1[(16 + lm) * LDK + kk];
    a1.y = h1[(16 + lm) * LDK + kk + 1];
    const float* wr = Wf2 + (size_t)kk * MM;
#pragma unroll
    for (int n = 0; n < 8; ++n) {
      v2f b;
      b.x = wr[n * 16 + lm];
      b.y = wr[MM + n * 16 + lm];
      acc2[0][n] = WMMA_F32X4(a0, b, acc2[0][n]);
      acc2[1][n] = WMMA_F32X4(a1, b, acc2[1][n]);
    }
  }

  // ---- scatter-add messages to destinations ----
#pragma unroll
  for (int m = 0; m < 2; ++m) {
#pragma unroll
    for (int n = 0; n < 8; ++n) {
#pragma unroll
      for (int i = 0; i < 8; ++i) {
        const int d = dstl[m * 16 + i + 8 * half];
        unsafeAtomicAdd(summsg + (size_t)d * MM + n * 16 + lm, acc2[m][n][i]);
      }
    }
  }
}

// ---------------------------------------------------------------------------
// Kernel 2: fused LSTM step. gates = [hidden|x|summsg|h0] @ [W_ih;W_hh] + b.
// Each wave computes one 16-column slice of ALL FOUR gates (cols c, c+128,
// c+256, c+384) for 16 nodes -> only 4 accumulator tiles (32 VGPRs), and the
// sigmoid/tanh elementwise fuses straight out of registers (no gate scratch).
// 5000 waves total; N = 625*16 exactly, so no tail guards.
// ---------------------------------------------------------------------------
__global__ __launch_bounds__(64) void rrn_lstm(
    const float* __restrict__ hidden, const float* __restrict__ x,
    const float* __restrict__ summsg, const float* __restrict__ h0,
    const float* __restrict__ c0, const float* __restrict__ W_ih,
    const float* __restrict__ W_hh, const float* __restrict__ b_ih,
    const float* __restrict__ b_hh, float* __restrict__ outH,
    float* __restrict__ outC) {
  const int lane = threadIdx.x & 31;
  const int wave = threadIdx.x >> 5;
  const int half = lane >> 4;
  const int lm = lane & 15;

  const int w = blockIdx.x * 2 + wave;   // 0..4999
  const int nodeTile = w >> 3;           // 0..624
  const int cb = w & 7;                  // 16-col block within D=128
  const int node0 = nodeTile * 16;
  const int r = node0 + lm;
  const int col = cb * 16 + lm;

  v8f accI = splat8(b_ih[col] + b_hh[col]);
  v8f accF = splat8(b_ih[DD + col] + b_hh[DD + col]);
  v8f accG = splat8(b_ih[2 * DD + col] + b_hh[2 * DD + col]);
  v8f accO = splat8(b_ih[3 * DD + col] + b_hh[3 * DD + col]);

  auto kblock = [&](const float* __restrict__ abase, const float* __restrict__ wbase) {
#pragma unroll 1
    for (int k = 0; k < 128; k += 4) {
      const int kk = k + 2 * half;
      v2f a;
      a.x = abase[kk];
      a.y = abase[kk + 1];
      const float* wr = wbase + (size_t)kk * (4 * DD) + cb * 16 + lm;
      v2f b;
      b.x = wr[0];            b.y = wr[4 * DD];            accI = WMMA_F32X4(a, b, accI);
      b.x = wr[DD];           b.y = wr[4 * DD + DD];       accF = WMMA_F32X4(a, b, accF);
      b.x = wr[2 * DD];       b.y = wr[4 * DD + 2 * DD];   accG = WMMA_F32X4(a, b, accG);
      b.x = wr[3 * DD];       b.y = wr[4 * DD + 3 * DD];   accO = WMMA_F32X4(a, b, accO);
    }
  };

  kblock(hidden + (size_t)r * DD, W_ih);                         // K   0..127
  kblock(x + (size_t)r * DD, W_ih + (size_t)128 * 4 * DD);       // K 128..255
  kblock(summsg + (size_t)r * DD, W_ih + (size_t)256 * 4 * DD);  // K 256..383
  kblock(h0 + (size_t)r * DD, W_hh);                             // K 384..511 (h0 @ W_hh)

  // elementwise LSTM nonlinearity, straight out of registers
#pragma unroll
  for (int i = 0; i < 8; ++i) {
    const int node = node0 + i + 8 * half;
    const float ig = accI[i];
    const float fg = accF[i];
    const float gg = accG[i];
    const float og = accO[i];
    const float c0v = c0[(size_t)node * DD + col];
    const float cn = sigmoid_(fg) * c0v + sigmoid_(ig) * tanhf(gg);
    const float hn = sigmoid_(og) * tanhf(cn);
    outH[(size_t)node * DD + col] = hn;
    outC[(size_t)node * DD + col] = cn;
  }
}

// ---------------------------------------------------------------------------
// Kernel 3: output MLP  relu(h_new @ Wo1 + bo1) @ Wo2 + bo2
// ---------------------------------------------------------------------------
__global__ __launch_bounds__(64) void rrn_out_mlp(
    const float* __restrict__ hN, const float* __restrict__ Wo1,
    const float* __restrict__ bo1, const float* __restrict__ Wo2,
    const float* __restrict__ bo2, float* __restrict__ out) {
  constexpr int LDK = 132;                 // 128 + 4 pad
  __shared__ float h1[2][16 * LDK];

  const int lane = threadIdx.x & 31;
  const int wave = threadIdx.x >> 5;
  const int half = lane >> 4;
  const int lm = lane & 15;
  const int node0 = (blockIdx.x * 2 + wave) * 16;
  if (node0 >= NN) return;
  const int r = min(node0 + lm, NN - 1);
  const float* arow = hN + (size_t)r * DD;

  v8f acc[8];
#pragma unroll
  for (int n = 0; n < 8; ++n) acc[n] = splat8(bo1[n * 16 + lm]);

#pragma unroll 1
  for (int k = 0; k < 128; k += 4) {
    const int kk = k + 2 * half;
    v2f a;
    a.x = arow[kk];
    a.y = arow[kk + 1];
    const float* wr = Wo1 + (size_t)kk * OHID;
#pragma unroll
    for (int n = 0; n < 8; ++n) {
      v2f b;
      b.x = wr[n * 16 + lm];
      b.y = wr[OHID + n * 16 + lm];
      acc[n] = WMMA_F32X4(a, b, acc[n]);
    }
  }

#pragma unroll
  for (int n = 0; n < 8; ++n) {
#pragma unroll
    for (int i = 0; i < 8; ++i) {
      h1[wave][(i + 8 * half) * LDK + n * 16 + lm] = fmaxf(acc[n][i], 0.0f);
    }
  }

  v8f acc2[4];
#pragma unroll
  for (int n = 0; n < 4; ++n) acc2[n] = splat8(bo2[n * 16 + lm]);

#pragma unroll 1
  for (int k = 0; k < 128; k += 4) {
    const int kk = k + 2 * half;
    v2f a;
    a.x = h1[wave][lm * LDK + kk];
    a.y = h1[wave][lm * LDK + kk + 1];
    const float* wr = Wo2 + (size_t)kk * OOUT;
#pragma unroll
    for (int n = 0; n < 4; ++n) {
      v2f b;
      b.x = wr[n * 16 + lm];
      b.y = wr[OOUT + n * 16 + lm];
      acc2[n] = WMMA_F32X4(a, b, acc2[n]);
    }
  }

#pragma unroll
  for (int n = 0; n < 4; ++n) {
#pragma unroll
    for (int i = 0; i < 8; ++i) {
      const int node = node0 + i + 8 * half;
      if (node < NN) out[(size_t)node * OOUT + n * 16 + lm] = acc2[n][i];
    }
  }
}

// ---------------------------------------------------------------------------
// Host launcher
// ---------------------------------------------------------------------------
extern "C" void kernel_launch(void* const* d_in, const int* in_sizes, int n_in,
                              void* d_out, int out_size, void* d_ws, size_t ws_size,
                              hipStream_t stream) {
  const float* x      = (const float*)d_in[0];
  const float* hidden = (const float*)d_in[1];
  const float* h0     = (const float*)d_in[2];
  const float* c0     = (const float*)d_in[3];
  const int*   src    = (const int*)d_in[4];
  const int*   dst    = (const int*)d_in[5];
  const float* Wf1    = (const float*)d_in[6];
  const float* bf1    = (const float*)d_in[7];
  const float* Wf2    = (const float*)d_in[8];
  const float* bf2    = (const float*)d_in[9];
  const float* W_ih   = (const float*)d_in[10];
  const float* W_hh   = (const float*)d_in[11];
  const float* b_ih   = (const float*)d_in[12];
  const float* b_hh   = (const float*)d_in[13];
  const float* Wo1    = (const float*)d_in[14];
  const float* bo1    = (const float*)d_in[15];
  const float* Wo2    = (const float*)d_in[16];
  const float* bo2    = (const float*)d_in[17];

  float* out  = (float*)d_out;                       // [N, 64]
  float* outH = out + (size_t)NN * OOUT;             // [N, 128]
  float* outC = outH + (size_t)NN * DD;              // [N, 128]
  float* summsg = (float*)d_ws;                      // [N, 128] scratch

  hipMemsetAsync(summsg, 0, (size_t)NN * MM * sizeof(float), stream);

  // 1 wave / block, 32 edges per wave
  rrn_edge_mlp<<<EE / 32, 32, 0, stream>>>(hidden, src, dst, Wf1, bf1, Wf2, bf2, summsg);
  // 2 waves / block, (16 nodes x 16 gate-cols x 4 gates) per wave: 5000 waves
  rrn_lstm<<<(NN / 16) * 8 / 2, 64, 0, stream>>>(hidden, x, summsg, h0, c0, W_ih, W_hh,
                                                 b_ih, b_hh, outH, outC);
  rrn_out_mlp<<<(NN + 31) / 32, 64, 0, stream>>>(outH, Wo1, bo1, Wo2, bo2, out);
}